// PNANetReverseMP_38053410243150
// MI455X (gfx1250) — compile-verified
//
#include <hip/hip_runtime.h>
#include <math.h>

// ---------------------------------------------------------------------------
// PNA GNN forward for MI455X (gfx1250, wave32).
//   * fp32 WMMA (v_wmma_f32_16x16x4_f32) for all dense transforms
//   * async global->LDS staging (ASYNCcnt) for shared A/B GEMM panels
//   * native f32 global atomics (add / min_num / max_num) for edge reductions
// ---------------------------------------------------------------------------

typedef __attribute__((ext_vector_type(2))) float v2f;
typedef __attribute__((ext_vector_type(8))) float v8f;

// ---- CDNA5 native f32 atomics (no-return, STOREcnt-tracked) ----------------
__device__ __forceinline__ void atom_add_f32(float* p, float v) {
  asm volatile("global_atomic_add_f32 %0, %1, off" :: "v"(p), "v"(v) : "memory");
}
__device__ __forceinline__ void atom_min_f32(float* p, float v) {
  asm volatile("global_atomic_min_num_f32 %0, %1, off" :: "v"(p), "v"(v) : "memory");
}
__device__ __forceinline__ void atom_max_f32(float* p, float v) {
  asm volatile("global_atomic_max_num_f32 %0, %1, off" :: "v"(p), "v"(v) : "memory");
}

// ---- async global -> LDS copy (16B per lane), ASYNCcnt-tracked -------------
// The low 32 bits of a generic pointer to LDS are the wave-relative LDS byte
// offset.  The generic pointer is also
// passed into the asm so the __shared__ allocation escapes and LLVM must
// treat the async copies as real writes to it (prevents undef-folding of the
// subsequent ds_loads).
__device__ __forceinline__ void async_copy_b128(float* lds, const float* g) {
  unsigned off = (unsigned)(unsigned long long)lds;
  asm volatile("global_load_async_to_lds_b128 %0, %1, off"
               :: "v"(off), "v"(g), "v"(lds) : "memory");
}
__device__ __forceinline__ void wait_async_all() {
#if defined(__gfx1250__) && __has_builtin(__builtin_amdgcn_s_wait_asynccnt)
  __builtin_amdgcn_s_wait_asynccnt(0);
#else
  asm volatile("s_wait_asynccnt 0x0" ::: "memory");
#endif
}

// ---- WMMA f32 16x16x4 micro-op --------------------------------------------
__device__ __forceinline__ v8f wmma_f32_16x16x4(v2f a, v2f b, v8f c) {
#if defined(__gfx1250__) && __has_builtin(__builtin_amdgcn_wmma_f32_16x16x4_f32)
  return __builtin_amdgcn_wmma_f32_16x16x4_f32(false, a, false, b, (short)0, c,
                                               false, false);
#else
  c[0] += a.x * b.x;  // host-pass placeholder
  return c;
#endif
}

// ---------------------------------------------------------------------------
// Cooperative WMMA GEMM: block = 4 waves = one 16-row M-tile x 64 columns.
// B panel (K x 64) and the shared A tile (16 x K) are async-staged into LDS;
// WMMA operands come from ds_load. Requires M%16==0, Ncols%64==0, K%4==0.
// ---------------------------------------------------------------------------
__global__ void gemm16(const float* __restrict__ A, int lda,
                       const float* __restrict__ B, int ldb,
                       float* __restrict__ C, int ldc,
                       int M, int K, int Ncols,
                       const float* __restrict__ bias, int accum, int do_relu) {
  __shared__ float smem[64 * 64 + 16 * 64];  // Bs[K][64] | As[16][K]
  const int AS_BASE = 64 * 64;               // float index of As

  int ngrp = Ncols >> 6;                 // groups of 64 columns
  int tm = blockIdx.x / ngrp;            // block-uniform
  int tg = blockIdx.x - tm * ngrp;
  int tid = threadIdx.y * 32 + threadIdx.x;

  // stage B[0..K)[tg*64 .. tg*64+64) -> Bs (compacted to width 64)
  int bsegs = K << 4;  // K rows * 16 segments of 4 floats
  for (int s = tid; s < bsegs; s += 128) {
    int row = s >> 4, c4 = (s & 15) << 2;
    async_copy_b128(smem + row * 64 + c4,
                    B + (size_t)row * ldb + tg * 64 + c4);
  }
  // stage A[tm*16 .. +16)[0..K) -> As (shared by all 4 waves)
  int kq = K >> 2;
  int asegs = kq << 4;  // 16 rows * K/4 segments
  for (int s = tid; s < asegs; s += 128) {
    int row = s / kq, kk = (s - row * kq) << 2;
    async_copy_b128(smem + AS_BASE + row * K + kk,
                    A + (size_t)(tm * 16 + row) * lda + kk);
  }
  wait_async_all();
  __syncthreads();

  int lane = threadIdx.x, half = lane >> 4, l16 = lane & 15;
  int wy = threadIdx.y;  // which 16-column sub-tile
  const float* As = smem + AS_BASE + (size_t)l16 * K + 2 * half;
  const float* Bs = smem + (size_t)(2 * half) * 64 + wy * 16 + l16;

  v8f acc = {0.f, 0.f, 0.f, 0.f, 0.f, 0.f, 0.f, 0.f};
  for (int k = 0; k < K; k += 4) {
    v2f a, b;
    a.x = As[k];
    a.y = As[k + 1];
    b.x = Bs[k * 64];
    b.y = Bs[(k + 1) * 64];
    acc = wmma_f32_16x16x4(a, b, acc);
  }

  int col = tg * 64 + wy * 16 + l16;
  float bv = bias ? bias[col] : 0.0f;
#pragma unroll
  for (int r = 0; r < 8; ++r) {
    int row = tm * 16 + r + 8 * half;
    size_t ci = (size_t)row * ldc + col;
    float v = acc[r] + bv;
    if (accum) v += C[ci];
    if (do_relu) v = fmaxf(v, 0.0f);
    C[ci] = v;
  }
}

// ---- degree counting --------------------------------------------------------
__global__ void deg_kernel(const int* __restrict__ src, const int* __restrict__ dst,
                           float* deg_f, float* deg_r, int E) {
  int e = blockIdx.x * blockDim.x + threadIdx.x;
  if (e >= E) return;
  atom_add_f32(&deg_f[dst[e]], 1.0f);
  atom_add_f32(&deg_r[src[e]], 1.0f);
}

// ---- sum of log(deg+1) for both directions ---------------------------------
__global__ void logsum_kernel(const float* __restrict__ deg_f,
                              const float* __restrict__ deg_r,
                              float* scal, int n) {
  __shared__ float sf[256], sr[256];
  int tid = threadIdx.x;
  int i = blockIdx.x * blockDim.x + tid;
  float a = 0.f, b = 0.f;
  if (i < n) {
    a = logf(deg_f[i] + 1.0f);
    b = logf(deg_r[i] + 1.0f);
  }
  sf[tid] = a; sr[tid] = b;
  __syncthreads();
  for (int s = 128; s > 0; s >>= 1) {
    if (tid < s) { sf[tid] += sf[tid + s]; sr[tid] += sr[tid + s]; }
    __syncthreads();
  }
  if (tid == 0) {
    atom_add_f32(&scal[0], sf[0]);
    atom_add_f32(&scal[1], sr[0]);
  }
}

// ---- repack pre-MLP weights: Wpre[k][c] (c<256: dst half + bias, else src) -
__global__ void repack_pre(const float* __restrict__ pW,   // [T,128,64] slice
                           const float* __restrict__ pb,   // [T,64] slice
                           float* __restrict__ Wpre,       // [64,512]
                           float* __restrict__ biasPre) {  // [512]
  int idx = blockIdx.x * blockDim.x + threadIdx.x;
  if (idx >= 64 * 512) return;
  int k = idx >> 9, c = idx & 511;
  int t = (c & 255) >> 6, f = c & 63;
  int kk = (c < 256) ? k : (k + 64);
  Wpre[k * 512 + c] = pW[(t * 128 + kk) * 64 + f];
  if (k == 0) biasPre[c] = (c < 256) ? pb[t * 64 + f] : 0.0f;
}

__global__ void fill_minmax(float* __restrict__ mn, float* __restrict__ mx, int total) {
  int idx = blockIdx.x * blockDim.x + threadIdx.x;
  if (idx >= total) return;
  mn[idx] = 3.402823466e38f;
  mx[idx] = -3.402823466e38f;
}

// ---- edge scatter: msg = preA[d] + preB[s]; segment sum/sq/min/max on d ----
__global__ void edge_agg(const int* __restrict__ sidx, const int* __restrict__ didx,
                         const float* __restrict__ pre,  // [N,512]
                         float* __restrict__ sum, float* __restrict__ sumsq,
                         float* __restrict__ mn, float* __restrict__ mx,
                         int total) {
  int idx = blockIdx.x * blockDim.x + threadIdx.x;
  if (idx >= total) return;
  int c = idx & 255, e = idx >> 8;
  __builtin_prefetch(sidx + e + 4096, 0, 1);  // global_prefetch on index stream
  int s = sidx[e], d = didx[e];
  float m = pre[(size_t)d * 512 + c] + pre[(size_t)s * 512 + 256 + c];
  size_t o = (size_t)d * 256 + c;
  atom_add_f32(&sum[o], m);
  atom_add_f32(&sumsq[o], m * m);
  atom_min_f32(&mn[o], m);
  atom_max_f32(&mx[o], m);
}

// ---- finalize: agg[n][t*256 + {mean,min,max,std}*64 + f] -------------------
__global__ void finalize_agg(const float* __restrict__ sum, const float* __restrict__ sumsq,
                             const float* __restrict__ mn, const float* __restrict__ mx,
                             const float* __restrict__ deg,
                             float* __restrict__ agg, int Nn) {
  int idx = blockIdx.x * blockDim.x + threadIdx.x;
  if (idx >= Nn * 256) return;
  int node = idx >> 8, c = idx & 255;
  int t = c >> 6, f = c & 63;
  float dg = deg[node];
  float d1 = fmaxf(dg, 1.0f);
  float mean = sum[idx] / d1;
  float mean2 = sumsq[idx] / d1;
  float var = mean2 - mean * mean;
  if (var < 0.f) var = 0.f;
  float sd = sqrtf(var + 1e-5f);
  float mnv = mn[idx], mxv = mx[idx];
  if (dg <= 0.f) { mnv = 0.f; mxv = 0.f; }
  size_t base = (size_t)node * 1024 + t * 256 + f;
  agg[base] = mean;
  agg[base + 64] = mnv;
  agg[base + 128] = mxv;
  agg[base + 192] = sd;
}

// ---- post MLP: hs = h@Wx + s1*(agg@Wamp) + s2*(agg@Watt) + agg@Wid + qb ----
// Block = (tower, 4 node-tiles). The full 832x16 tower weight panel is
// async-staged to LDS once per block and shared by the 4 waves; 4 WMMA
// accumulator chains per wave. Out-of-range tiles clamp (EXEC stays full).
__global__ void post_kernel(const float* __restrict__ h, const float* __restrict__ agg,
                            const float* __restrict__ qW,  // [T,832,16] slice
                            const float* __restrict__ qb,  // [T,16] slice
                            const float* __restrict__ deg,
                            const float* __restrict__ sumlog,
                            float* __restrict__ hs, int Nn) {
  __shared__ float smem[832 * 16];  // 52 KB tower weight panel
  int mtiles = Nn >> 4;
  int t = blockIdx.x & 3;            // tower (block-uniform)
  int tmg = blockIdx.x >> 2;         // node-tile group
  int tid = threadIdx.y * 32 + threadIdx.x;

  const float* W = qW + (size_t)t * 832 * 16;
  for (int s = tid; s < (832 * 16) / 4; s += 128)  // 3328 x b128 segments
    async_copy_b128(smem + s * 4, W + s * 4);
  wait_async_all();
  __syncthreads();

  int tm = tmg * 4 + threadIdx.y;          // wave-uniform
  int tmc = (tm < mtiles) ? tm : (mtiles - 1);  // clamp; skip store if OOB
  int lane = threadIdx.x, half = lane >> 4, l16 = lane & 15;

  const float* Ah = h + (size_t)(tmc * 16 + l16) * 64 + 2 * half;
  const float* Ag = agg + (size_t)(tmc * 16 + l16) * 1024 + t * 256 + 2 * half;
  const float* Bx = smem + (size_t)(0 + 2 * half) * 16 + l16;
  const float* B1 = smem + (size_t)(64 + 2 * half) * 16 + l16;   // amplification
  const float* B2 = smem + (size_t)(320 + 2 * half) * 16 + l16;  // attenuation
  const float* B3 = smem + (size_t)(576 + 2 * half) * 16 + l16;  // identity

  v8f ax = {0.f, 0.f, 0.f, 0.f, 0.f, 0.f, 0.f, 0.f};
  v8f a1 = ax, a2 = ax, a3 = ax;

  for (int k = 0; k < 64; k += 4) {
    v2f a, b;
    a.x = Ah[k]; a.y = Ah[k + 1];
    b.x = Bx[k * 16]; b.y = Bx[(k + 1) * 16];
    ax = wmma_f32_16x16x4(a, b, ax);
  }
  for (int k = 0; k < 256; k += 4) {
    v2f a, b;
    a.x = Ag[k]; a.y = Ag[k + 1];
    b.x = B1[k * 16]; b.y = B1[(k + 1) * 16];
    a1 = wmma_f32_16x16x4(a, b, a1);
    b.x = B2[k * 16]; b.y = B2[(k + 1) * 16];
    a2 = wmma_f32_16x16x4(a, b, a2);
    b.x = B3[k * 16]; b.y = B3[(k + 1) * 16];
    a3 = wmma_f32_16x16x4(a, b, a3);
  }

  if (tm < mtiles) {  // wave-uniform store guard
    float avg = sumlog[0] / (float)Nn;
    float qbv = qb[t * 16 + l16];
#pragma unroll
    for (int r = 0; r < 8; ++r) {
      int row = tm * 16 + r + 8 * half;
      float dg = deg[row];
      float logd = logf(fmaxf(dg, 1.0f) + 1.0f);
      float s1 = logd / avg, s2 = avg / logd;
      hs[(size_t)row * 64 + t * 16 + l16] =
          ax[r] + s1 * a1[r] + s2 * a2[r] + a3[r] + qbv;
    }
  }
}

// ---- batchnorm stats (per column over N rows) ------------------------------
__global__ void bn_stats(const float* __restrict__ o, float* mu, float* var, int Nn) {
  int c = blockIdx.x;  // 64 columns
  int tid = threadIdx.x;
  float s = 0.f, s2 = 0.f;
  for (int n = tid; n < Nn; n += blockDim.x) {
    float v = o[(size_t)n * 64 + c];
    s += v;
    s2 += v * v;
  }
  __shared__ float ss[256], ssq[256];
  ss[tid] = s; ssq[tid] = s2;
  __syncthreads();
  for (int st = 128; st > 0; st >>= 1) {
    if (tid < st) { ss[tid] += ss[tid + st]; ssq[tid] += ssq[tid + st]; }
    __syncthreads();
  }
  if (tid == 0) {
    float m = ss[0] / (float)Nn;
    mu[c] = m;
    var[c] = ssq[0] / (float)Nn - m * m;
  }
}

__global__ void bn_apply(const float* __restrict__ o, const float* __restrict__ mu,
                         const float* __restrict__ var, const float* __restrict__ g,
                         const float* __restrict__ b, float* __restrict__ h, int total) {
  int idx = blockIdx.x * blockDim.x + threadIdx.x;
  if (idx >= total) return;
  int c = idx & 63;
  float v = g[c] * (o[idx] - mu[c]) * rsqrtf(var[c] + 1e-5f) + b[c];
  h[idx] = fmaxf(v, 0.0f);
}

// ---- final projection [N,64] @ [64,2] + b ----------------------------------
__global__ void final_out(const float* __restrict__ t1, const float* __restrict__ W2,
                          const float* __restrict__ b2, float* __restrict__ out, int Nn) {
  int idx = blockIdx.x * blockDim.x + threadIdx.x;
  if (idx >= Nn * 2) return;
  int node = idx >> 1, o = idx & 1;
  float s = b2[o];
  const float* row = t1 + (size_t)node * 64;
#pragma unroll 16
  for (int k = 0; k < 64; ++k) s = fmaf(row[k], W2[k * 2 + o], s);
  out[idx] = s;
}

// ---------------------------------------------------------------------------
extern "C" void kernel_launch(void* const* d_in, const int* in_sizes, int n_in,
                              void* d_out, int out_size, void* d_ws, size_t ws_size,
                              hipStream_t stream) {
  const float* x      = (const float*)d_in[0];
  const int*   ei     = (const int*)  d_in[1];
  const float* in_W   = (const float*)d_in[2];
  const float* in_b   = (const float*)d_in[3];
  const float* pre_W  = (const float*)d_in[4];
  const float* pre_b  = (const float*)d_in[5];
  const float* post_W = (const float*)d_in[6];
  const float* post_b = (const float*)d_in[7];
  const float* lin_W  = (const float*)d_in[8];
  const float* lin_b  = (const float*)d_in[9];
  const float* bn_g   = (const float*)d_in[10];
  const float* bn_b   = (const float*)d_in[11];
  const float* W1     = (const float*)d_in[12];
  const float* b1     = (const float*)d_in[13];
  const float* W2     = (const float*)d_in[14];
  const float* b2     = (const float*)d_in[15];

  const int Nn = in_sizes[0] / 32;  // 20000
  const int E  = in_sizes[1] / 2;   // 120000
  const int L  = 2;

  // ---- workspace layout (floats) ----
  float* ws = (float*)d_ws;
  size_t p = 0;
  float* h       = ws + p; p += (size_t)Nn * 64;
  float* obuf    = ws + p; p += (size_t)Nn * 64;
  float* hsf     = ws + p; p += (size_t)Nn * 64;
  float* pre     = ws + p; p += (size_t)Nn * 512;
  float* agg     = ws + p; p += (size_t)Nn * 1024;
  float* sumb    = ws + p; p += (size_t)Nn * 256;
  float* sqb     = ws + p; p += (size_t)Nn * 256;   // contiguous after sumb
  float* mnb     = ws + p; p += (size_t)Nn * 256;
  float* mxb     = ws + p; p += (size_t)Nn * 256;
  float* deg_f   = ws + p; p += (size_t)Nn;
  float* deg_r   = ws + p; p += (size_t)Nn;         // contiguous after deg_f
  float* scal    = ws + p; p += 4;                  // contiguous after deg_r
  float* Wpre    = ws + p; p += 64 * 512;
  float* biasPre = ws + p; p += 512;
  float* mu      = ws + p; p += 64;
  float* var     = ws + p; p += 64;

  auto launch_gemm = [&](const float* A, int lda, const float* B, int ldb,
                         float* C, int ldc, int M, int K, int Ncols,
                         const float* bias, int accum, int relu) {
    int grid = (M / 16) * (Ncols / 64);
    hipLaunchKernelGGL(gemm16, dim3(grid), dim3(32, 4), 0, stream,
                       A, lda, B, ldb, C, ldc, M, K, Ncols, bias, accum, relu);
  };

  // degrees + avg log-degree scalars
  hipMemsetAsync(deg_f, 0, (size_t)(2 * Nn + 4) * sizeof(float), stream);
  hipLaunchKernelGGL(deg_kernel, dim3((E + 255) / 256), dim3(256), 0, stream,
                     ei, ei + E, deg_f, deg_r, E);
  hipLaunchKernelGGL(logsum_kernel, dim3((Nn + 255) / 256), dim3(256), 0, stream,
                     deg_f, deg_r, scal, Nn);

  // input layer: h = relu(x @ in_W + in_b)
  launch_gemm(x, 32, in_W, 64, h, 64, Nn, 32, 64, in_b, 0, 1);

  for (int l = 0; l < L; ++l) {
    for (int dd = 0; dd < 2; ++dd) {
      const int slot = l * 2 + dd;
      const int* sidx = (dd == 0) ? ei : ei + E;       // source of messages
      const int* didx = (dd == 0) ? ei + E : ei;       // aggregation key
      const float* degp = (dd == 0) ? deg_f : deg_r;

      // repack pre-MLP weights for this (layer,dir); bias folded into dst half
      hipLaunchKernelGGL(repack_pre, dim3((64 * 512) / 256), dim3(256), 0, stream,
                         pre_W + (size_t)slot * 4 * 128 * 64,
                         pre_b + (size_t)slot * 4 * 64, Wpre, biasPre);

      // per-node pre transform: pre[N,512] = h @ Wpre + biasPre
      launch_gemm(h, 64, Wpre, 512, pre, 512, Nn, 64, 512, biasPre, 0, 0);

      // init segment accumulators
      hipMemsetAsync(sumb, 0, (size_t)2 * Nn * 256 * sizeof(float), stream);
      hipLaunchKernelGGL(fill_minmax, dim3((Nn * 256 + 255) / 256), dim3(256), 0,
                         stream, mnb, mxb, Nn * 256);

      // edge scatter with native f32 atomics (L2-resident)
      int etotal = E * 256;
      hipLaunchKernelGGL(edge_agg, dim3((etotal + 255) / 256), dim3(256), 0, stream,
                         sidx, didx, pre, sumb, sqb, mnb, mxb, etotal);

      // mean/min/max/std -> agg [N,1024]
      hipLaunchKernelGGL(finalize_agg, dim3((Nn * 256 + 255) / 256), dim3(256), 0,
                         stream, sumb, sqb, mnb, mxb, degp, agg, Nn);

      // post towers -> hsf [N,64]  (grid = towers * ceil(mtiles/4))
      int mtiles = Nn / 16;
      hipLaunchKernelGGL(post_kernel, dim3(((mtiles + 3) / 4) * 4), dim3(32, 4), 0,
                         stream, h, agg,
                         post_W + (size_t)slot * 4 * 832 * 16,
                         post_b + (size_t)slot * 4 * 16,
                         degp, scal + dd, hsf, Nn);

      // lin: obuf (+)= hsf @ lin_W + lin_b   (accumulate 2nd direction)
      launch_gemm(hsf, 64, lin_W + (size_t)slot * 64 * 64, 64, obuf, 64,
                  Nn, 64, 64, lin_b + (size_t)slot * 64, dd, 0);
    }

    // batchnorm + relu -> h
    hipLaunchKernelGGL(bn_stats, dim3(64), dim3(256), 0, stream, obuf, mu, var, Nn);
    hipLaunchKernelGGL(bn_apply, dim3((Nn * 64 + 255) / 256), dim3(256), 0, stream,
                       obuf, mu, var, bn_g + l * 64, bn_b + l * 64, h, Nn * 64);
  }

  // MLP: t1 = relu(h @ W1 + b1);  out = t1 @ W2 + b2
  launch_gemm(h, 64, W1, 64, hsf, 64, Nn, 64, 64, b1, 0, 1);
  hipLaunchKernelGGL(final_out, dim3((Nn * 2 + 255) / 256), dim3(256), 0, stream,
                     hsf, W2, b2, (float*)d_out, Nn);
}